// GridPEAttention_61512521613905
// MI455X (gfx1250) — compile-verified
//
#include <hip/hip_runtime.h>
#include <hip/hip_bf16.h>

// ---------------------------------------------------------------------------
// GridPE attention for MI455X (gfx1250), fp32 throughout via
// V_WMMA_F32_16X16X4_F32 (fp32 A/B, fp32 accumulate) to match the fp32
// reference numerics while running on the matrix pipes. GEMMs stage panels
// through LDS with the Tensor Data Mover (TENSOR_LOAD_TO_LDS + TENSORcnt),
// double-buffered one K-chunk ahead.
//
// Shapes (fixed): B=8, N=1025, C=768, H=8, hd=96, n=2, S=24, D=2
// ---------------------------------------------------------------------------

typedef __attribute__((ext_vector_type(2))) float        v2f;
typedef __attribute__((ext_vector_type(8))) float        v8f;
typedef __attribute__((ext_vector_type(4))) unsigned int v4u;
typedef __attribute__((ext_vector_type(4))) int          v4i;
typedef __attribute__((ext_vector_type(8))) int          v8i;

#define BB   8
#define NN   1025
#define CC   768
#define HH   8
#define HD   96
#define NROT 2
#define SS   24

#if defined(__has_builtin)
#if __has_builtin(__builtin_amdgcn_tensor_load_to_lds) && \
    __has_builtin(__builtin_amdgcn_s_wait_tensorcnt)
#define HAVE_TDM 1
#endif
#endif

#ifdef HAVE_TDM
// ---------------------------------------------------------------------------
// Issue one TDM 2-D tile load (global -> LDS), fp32 elements.
// tile_d0 contiguous elements per row, tile_d1 rows; rows land in LDS at a
// stride of (tile_d0 + 2) dwords via the TDM pad feature (pad every 32 dwords
// by 2 dwords) so downstream ds_load_b64 fragments are bank-conflict free.
// tensor_d0/d1 are the remaining extents from the tile origin: TDM zero-fills
// OOB reads, which handles the M=8200 ragged edge for free.
// ---------------------------------------------------------------------------
__device__ __forceinline__ void tdm_load_2d_f32(
    unsigned lds_byte_addr, const float* gaddr,
    unsigned tile_d0, unsigned tile_d1,
    unsigned tensor_d0, unsigned tensor_d1,
    unsigned stride0_elems)
{
  unsigned long long ga = (unsigned long long)(const void*)gaddr;
  v4u g0 = {
    1u,                                                  // count=1, user mode
    lds_byte_addr,                                       // LDS dest (bytes)
    (unsigned)ga,                                        // global_addr[31:0]
    (unsigned)((ga >> 32) & 0x01FFFFFFu) | (2u << 30)    // addr[56:32] | type=2
  };
  v8i g1 = {
    (int)((2u << 16)                                     // data_size = 4B
        | (1u << 20)                                     // pad_enable
        | (4u << 22)                                     // pad_interval: 32 dwords
        | (1u << 25)),                                   // pad_amount: 2 dwords
    (int)((tensor_d0 & 0xFFFFu) << 16),                  // tensor_dim0[15:0]
    (int)((tensor_d0 >> 16) | ((tensor_d1 & 0xFFFFu) << 16)),
    (int)((tensor_d1 >> 16) | (tile_d0 << 16)),          // dim1 hi | tile_dim0
    (int)(tile_d1 & 0xFFFFu),                            // tile_dim1 (tile_dim2=0)
    (int)stride0_elems,                                  // tensor_dim0_stride lo32
    0, 0                                                 // stride hi | dim1_stride=0
  };
  v4i gz = {0, 0, 0, 0};
#if __clang_major__ >= 23
  v8i gz8 = {0, 0, 0, 0, 0, 0, 0, 0};
  __builtin_amdgcn_tensor_load_to_lds(g0, g1, gz, gz, gz8, 0);
#else
  __builtin_amdgcn_tensor_load_to_lds(g0, g1, gz, gz, 0);
#endif
}
#endif // HAVE_TDM

// ---------------------------------------------------------------------------
// Kernel 1/4: tiled GEMM  C[M,Nc] = A[M,K] @ B[K,Nc] (+bias), fp32 WMMA.
// Block: 256 thr = 8 waves as 4(M) x 2(N); block tile 64x32; K chunk 32.
// TDM double-buffers A(64x32) and B(32x32) panels in LDS, one chunk ahead.
// WMMA fragment layouts (ISA 7.12.2, f32):
//   A 16x4:  lane&15 = M row, (lane>>4)*2 + vgpr = K      (ds_load_b64)
//   B 4x16:  lane&15 = N col, (lane>>4)*2 + vgpr = K      (2x ds_load_b32)
//   C/D:     vgpr i -> M = i + 8*(lane>=16), lane&15 = N
// ---------------------------------------------------------------------------
#define LDA 34   // 32 data + 2 pad dwords per row (8B-aligned, conflict-free)

template<bool ADD_BIAS>
__global__ __launch_bounds__(256) void gemm_wmma_f32(
    const float* __restrict__ A, const float* __restrict__ Bm,
    const float* __restrict__ bias, float* __restrict__ Cm,
    int M, int Nc, int K)
{
  __shared__ float sA[2][64][LDA];
  __shared__ float sB[2][32][LDA];

  const int lane  = threadIdx.x & 31;
  const int wave  = threadIdx.x >> 5;
  const int wm    = wave >> 1;             // 0..3
  const int wn    = wave & 1;              // 0..1
  const int m0    = blockIdx.y * 64;
  const int n0    = blockIdx.x * 32;
  const int mcol  = lane & 15;
  const int khalf = (lane >> 4) << 1;      // 0 or 2

  v8f acc = {};
  const int nChunks = K >> 5;

#ifdef HAVE_TDM
  if (wave == 0) {
    tdm_load_2d_f32((unsigned)(size_t)&sA[0][0][0], A + (size_t)m0 * K,
                    32, 64, (unsigned)K, (unsigned)(M - m0), (unsigned)K);
    tdm_load_2d_f32((unsigned)(size_t)&sB[0][0][0], Bm + n0,
                    32, 32, (unsigned)(Nc - n0), (unsigned)K, (unsigned)Nc);
  }
  for (int chunk = 0; chunk < nChunks; ++chunk) {
    const int buf = chunk & 1;
    if (wave == 0) {
      if (chunk + 1 < nChunks) {
        const int k1 = (chunk + 1) << 5;
        tdm_load_2d_f32((unsigned)(size_t)&sA[buf ^ 1][0][0],
                        A + (size_t)m0 * K + k1,
                        32, 64, (unsigned)(K - k1), (unsigned)(M - m0),
                        (unsigned)K);
        tdm_load_2d_f32((unsigned)(size_t)&sB[buf ^ 1][0][0],
                        Bm + (size_t)k1 * Nc + n0,
                        32, 32, (unsigned)(Nc - n0), (unsigned)(K - k1),
                        (unsigned)Nc);
        __builtin_amdgcn_s_wait_tensorcnt(2);  // current chunk's pair done
      } else {
        __builtin_amdgcn_s_wait_tensorcnt(0);
      }
    }
    __syncthreads();
    #pragma unroll
    for (int kk = 0; kk < 8; ++kk) {
      v2f a = *(const v2f*)&sA[buf][wm * 16 + mcol][kk * 4 + khalf];
      v2f b;
      b[0] = sB[buf][kk * 4 + khalf][wn * 16 + mcol];
      b[1] = sB[buf][kk * 4 + khalf + 1][wn * 16 + mcol];
      acc = __builtin_amdgcn_wmma_f32_16x16x4_f32(
          false, a, false, b, (short)0, acc, false, false);
    }
    __syncthreads();   // buf^1 safe to refill next iteration
  }
#else
  // Fallback: cooperative synchronous staging (no TDM builtin available).
  for (int chunk = 0; chunk < nChunks; ++chunk) {
    const int k0 = chunk << 5;
    __syncthreads();
    for (int i = threadIdx.x; i < 64 * 32; i += 256) {
      int r = i >> 5, c = i & 31;
      int gr = m0 + r;
      sA[0][r][c] = (gr < M) ? A[(size_t)gr * K + k0 + c] : 0.0f;
    }
    for (int i = threadIdx.x; i < 32 * 32; i += 256) {
      int r = i >> 5, c = i & 31;
      sB[0][r][c] = Bm[(size_t)(k0 + r) * Nc + n0 + c];
    }
    __syncthreads();
    #pragma unroll
    for (int kk = 0; kk < 8; ++kk) {
      v2f a = *(const v2f*)&sA[0][wm * 16 + mcol][kk * 4 + khalf];
      v2f b;
      b[0] = sB[0][kk * 4 + khalf][wn * 16 + mcol];
      b[1] = sB[0][kk * 4 + khalf + 1][wn * 16 + mcol];
      acc = __builtin_amdgcn_wmma_f32_16x16x4_f32(
          false, a, false, b, (short)0, acc, false, false);
    }
  }
#endif

  const int rowhi = (lane >> 4) << 3;      // 0 or 8
  const int ncol  = n0 + wn * 16 + mcol;
  #pragma unroll
  for (int i = 0; i < 8; ++i) {
    int r = m0 + wm * 16 + rowhi + i;
    if (r < M) {
      float v = acc[i];
      if (ADD_BIAS) v += bias[ncol];
      Cm[(size_t)r * Nc + ncol] = v;
    }
  }
}

// ---------------------------------------------------------------------------
// Kernel 2/4: grid-RoPE rotation of q and k, in place inside qkv scratch.
// qkv layout: (B, N, 3, H, hd) row-major. Token 0 left unrotated.
// ---------------------------------------------------------------------------
__global__ __launch_bounds__(256) void rope_rotate(
    float* __restrict__ qkv, const float* __restrict__ positions,
    const float* __restrict__ omegas)
{
  long idx = (long)blockIdx.x * blockDim.x + threadIdx.x;  // B*H*(N-1)*n*S
  int s  = (int)(idx % SS);   idx /= SS;
  int nn = (int)(idx % NROT); idx /= NROT;
  int t1 = (int)(idx % (NN - 1)); idx /= (NN - 1);
  int h  = (int)(idx % HH);   idx /= HH;
  int b  = (int)idx;
  int t  = t1 + 1;

  float px = positions[((long)b * (NN - 1) + t1) * 2 + 0];
  float py = positions[((long)b * (NN - 1) + t1) * 2 + 1];
  float ox = omegas[(h * NROT + nn) * 2 + 0];
  float oy = omegas[(h * NROT + nn) * 2 + 1];
  float mag   = expf(-(float)s * (4.6051701859880914f / 24.0f)); // 100^(-s/24)
  float theta = (px * ox + py * oy) * mag;
  float sn = sinf(theta), cs = cosf(theta);

  int  d    = nn * (SS * 2) + s * 2;
  long base = (((long)b * NN + t) * 3) * CC + (long)h * HD + d;

  float qe = qkv[base],        qo = qkv[base + 1];
  qkv[base]     = qe * cs - qo * sn;
  qkv[base + 1] = qe * sn + qo * cs;

  long kb = base + CC;
  float ke = qkv[kb], ko = qkv[kb + 1];
  qkv[kb]     = ke * cs - ko * sn;
  qkv[kb + 1] = ke * sn + ko * cs;
}

// ---------------------------------------------------------------------------
// Kernel 3/4: flash attention, one (b, h, 16-query tile) per wave.
// S = Q K^T via 24 fp32 WMMAs; online softmax staged through per-wave LDS;
// O += P V via 24 fp32 WMMAs into 6 v8f accumulators (96 dims = 6x16).
// ---------------------------------------------------------------------------
__global__ __launch_bounds__(256) void attn_wmma_f32(
    const float* __restrict__ qkv, float* __restrict__ out)
{
  const int lane = threadIdx.x & 31;
  const int wave = threadIdx.x >> 5;

  long task = (long)blockIdx.x * 8 + wave;   // 64 * 65 = 4160 tasks, /8 exact
  const int QT = (NN + 15) / 16;             // 65
  int qtile = (int)(task % QT); task /= QT;
  int h = (int)(task % HH);
  int b = (int)(task / HH);

  __shared__ float sP[8][16][17];
  __shared__ float sAlpha[8][16];
  __shared__ float sL[8][16];

  const int mcol  = lane & 15;
  const int khalf = (lane >> 4) << 1;
  const int rowhi = (lane >> 4) << 3;

  int q_t = qtile * 16 + mcol; if (q_t > NN - 1) q_t = NN - 1;
  const float* qbase = qkv + (((long)b * NN + q_t) * 3) * CC + (long)h * HD;
  v2f qa[24];
  #pragma unroll
  for (int kk = 0; kk < 24; ++kk)
    qa[kk] = *(const v2f*)(qbase + kk * 4 + khalf);

  const float scale = 0.10206207261596577f;  // 96^-0.5
  float m_i = -3.0e38f, l_i = 0.0f;
  v8f o[6] = {};

  for (int kt = 0; kt < QT; ++kt) {
    const int kvalid = (kt == QT - 1) ? (NN - 16 * (QT - 1)) : 16;

    int key = kt * 16 + mcol; if (key > NN - 1) key = NN - 1;
    const float* kbase = qkv + (((long)b * NN + key) * 3 + 1) * CC + (long)h * HD;
    v8f s = {};
    #pragma unroll
    for (int kk = 0; kk < 24; ++kk) {
      v2f bf = *(const v2f*)(kbase + kk * 4 + khalf);
      s = __builtin_amdgcn_wmma_f32_16x16x4_f32(
          false, qa[kk], false, bf, (short)0, s, false, false);
    }
    #pragma unroll
    for (int i = 0; i < 8; ++i)
      sP[wave][rowhi + i][mcol] = s[i] * scale;
    __syncthreads();

    if (lane < 16) {
      float rmax = -3.0e38f;
      for (int c = 0; c < kvalid; ++c)
        rmax = fmaxf(rmax, sP[wave][lane][c]);
      float mnew  = fmaxf(m_i, rmax);
      float alpha = expf(m_i - mnew);
      float rsum  = 0.0f;
      for (int c = 0; c < 16; ++c) {
        float p = (c < kvalid) ? expf(sP[wave][lane][c] - mnew) : 0.0f;
        sP[wave][lane][c] = p;
        rsum += p;
      }
      l_i = l_i * alpha + rsum;
      m_i = mnew;
      sAlpha[wave][lane] = alpha;
    }
    __syncthreads();

    #pragma unroll
    for (int i = 0; i < 8; ++i) {
      float al = sAlpha[wave][rowhi + i];
      #pragma unroll
      for (int c = 0; c < 6; ++c) o[c][i] *= al;
    }

    #pragma unroll
    for (int kk = 0; kk < 4; ++kk) {
      v2f pa;
      pa[0] = sP[wave][mcol][kk * 4 + khalf];
      pa[1] = sP[wave][mcol][kk * 4 + khalf + 1];
      int k0 = kt * 16 + kk * 4 + khalf;
      int k0c = (k0     > NN - 1) ? NN - 1 : k0;
      int k1c = (k0 + 1 > NN - 1) ? NN - 1 : k0 + 1;
      const float* v0 = qkv + (((long)b * NN + k0c) * 3 + 2) * CC + (long)h * HD;
      const float* v1 = qkv + (((long)b * NN + k1c) * 3 + 2) * CC + (long)h * HD;
      #pragma unroll
      for (int c = 0; c < 6; ++c) {
        v2f bv;
        bv[0] = v0[c * 16 + mcol];
        bv[1] = v1[c * 16 + mcol];
        o[c] = __builtin_amdgcn_wmma_f32_16x16x4_f32(
            false, pa, false, bv, (short)0, o[c], false, false);
      }
    }
    __syncthreads();
  }

  if (lane < 16) sL[wave][lane] = l_i;
  __syncthreads();
  #pragma unroll
  for (int i = 0; i < 8; ++i) {
    int t = qtile * 16 + rowhi + i;
    if (t < NN) {
      float inv = 1.0f / sL[wave][rowhi + i];
      #pragma unroll
      for (int c = 0; c < 6; ++c)
        out[((long)b * NN + t) * CC + (long)h * HD + c * 16 + mcol] = o[c][i] * inv;
    }
  }
}

// ---------------------------------------------------------------------------
// Launch: qkv GEMM -> rope -> attention -> proj GEMM (+bias)
// Workspace: qkv scratch 8200*2304 f32 (75.6 MB) + attn 8200*768 f32 (25.2 MB)
// ---------------------------------------------------------------------------
extern "C" void kernel_launch(void* const* d_in, const int* in_sizes, int n_in,
                              void* d_out, int out_size, void* d_ws, size_t ws_size,
                              hipStream_t stream) {
  const float* x         = (const float*)d_in[0];
  const float* positions = (const float*)d_in[1];
  const float* Wqkv      = (const float*)d_in[2];
  const float* Wproj     = (const float*)d_in[3];
  const float* bproj     = (const float*)d_in[4];
  const float* omegas    = (const float*)d_in[5];
  float* out  = (float*)d_out;

  float* qkv  = (float*)d_ws;                          // (B*N, 3C)
  float* attn = qkv + (size_t)(BB * NN) * (3 * CC);    // (B*N, C)

  const int M = BB * NN;                               // 8200
  const int tilesM64 = (M + 63) / 64;                  // 129

  // qkv = x @ Wqkv : block tile 64x32 -> grid (2304/32, 129)
  gemm_wmma_f32<false><<<dim3(3 * CC / 32, tilesM64), 256, 0, stream>>>(
      x, Wqkv, nullptr, qkv, M, 3 * CC, CC);

  // B*H*(N-1)*n*S = 3,145,728 threads = 12288 blocks (exact)
  rope_rotate<<<12288, 256, 0, stream>>>(qkv, positions, omegas);

  // 8*8 heads * 65 query tiles = 4160 wave tasks = 520 blocks (exact)
  attn_wmma_f32<<<520, 256, 0, stream>>>(qkv, attn);

  // out = attn @ Wproj + bproj : grid (768/32, 129)
  gemm_wmma_f32<true><<<dim3(CC / 32, tilesM64), 256, 0, stream>>>(
      attn, Wproj, bproj, out, M, CC, CC);
}